// MPNNAttn_34643206210178
// MI455X (gfx1250) — compile-verified
//
#include <hip/hip_runtime.h>
#include <hip/hip_bf16.h>

typedef float v2f __attribute__((ext_vector_type(2)));
typedef float v8f __attribute__((ext_vector_type(8)));

#define BATCH 8
#define NNODE 512
#define BN    4096   // BATCH*NNODE
#define HID   64
#define HEADS 8

__device__ __forceinline__ float sigf(float x) { return 1.f / (1.f + __expf(-x)); }

// ---------------------------------------------------------------------------
// Generic fp32 WMMA GEMM: Y[m, ycol0 + z*yzcol + n] = act(X[m,:K] . W[n,:K] + bias[n])
// One wave (32 lanes) computes one 16x16 tile via V_WMMA_F32_16X16X4_F32.
// grid = (M/16, ceil(N/16), Z), block = 32.  W_eff = W + blockIdx.z * wzstride.
// act: 0=none 1=relu 2=sigmoid.  N need not be a multiple of 16 (stores guarded).
// ---------------------------------------------------------------------------
__global__ void k_gemm_wmma(const float* __restrict__ X, const float* __restrict__ W,
                            const float* __restrict__ bias, float* __restrict__ Y,
                            int M, int N, int K, int ldy, int ycol0,
                            long wzstride, int yzcol, int act)
{
    const int m0   = blockIdx.x * 16;
    const int n0   = blockIdx.y * 16;
    const float* Wp = W + (long)blockIdx.z * wzstride;
    const int ycb  = ycol0 + blockIdx.z * yzcol;

    const int lane = threadIdx.x;
    const int half = lane >> 4;       // 0: K={0,1}, 1: K={2,3} per ISA f32 A/B layout
    const int lr   = lane & 15;

    const int arow = m0 + lr;
    int wrow = n0 + lr; if (wrow > N - 1) wrow = N - 1;   // clamp for ragged N (e.g. 12)
    const float* xp = X  + (size_t)arow * K + 2 * half;
    const float* wp = Wp + (size_t)wrow * K + 2 * half;

    v8f acc = {};
    for (int k = 0; k < K; k += 4) {
        v2f a, b;
        a.x = xp[k]; a.y = xp[k + 1];
        b.x = wp[k]; b.y = wp[k + 1];
        acc = __builtin_amdgcn_wmma_f32_16x16x4_f32(false, a, false, b,
                                                    (short)0, acc, false, false);
    }

    // D layout: VGPR v -> row m0+v (lanes 0-15), m0+v+8 (lanes 16-31); col = n0+lr
    #pragma unroll
    for (int v = 0; v < 8; ++v) {
        int row = m0 + v + 8 * half;
        int col = n0 + lr;
        if (col < N) {
            float val = acc[v];
            if (bias) val += bias[col];
            if (act == 1)      val = val > 0.f ? val : 0.f;
            else if (act == 2) val = sigf(val);
            Y[(size_t)row * ldy + ycb + col] = val;
        }
    }
}

// ---------------------------------------------------------------------------
// h0 (zero-pad 50->64) and node mask
// ---------------------------------------------------------------------------
__global__ void k_h0(const float* __restrict__ h_in, float* __restrict__ h0,
                     float* __restrict__ mask)
{
    int id = blockIdx.x * 256 + threadIdx.x;
    if (id >= BN * HID) return;
    int f = id & 63, bn = id >> 6;
    h0[id] = (f < 50) ? h_in[bn * 50 + f] : 0.f;
    if (f == 0) {
        float s = 0.f;
        for (int k = 0; k < 50; ++k) s += h_in[bn * 50 + k];
        mask[bn] = (s > 0.f) ? 1.f : 0.f;
    }
}

// ---------------------------------------------------------------------------
// s_i / s_j: 64-dim dots of z with attn_w slices (w_j = [:64], w_i = [64:128])
// one thread per (b,h,n), id = (b*8+h)*512 + n
// ---------------------------------------------------------------------------
__global__ void k_sij(const float* __restrict__ Z, const float* __restrict__ aw,
                      float* __restrict__ si, float* __restrict__ sj)
{
    int id = blockIdx.x * 256 + threadIdx.x;
    if (id >= BATCH * HEADS * NNODE) return;
    int n = id & 511, h = (id >> 9) & 7, b = id >> 12;
    const float* zp = Z + ((size_t)(b * NNODE + n)) * 512 + h * 64;
    const float* wj = aw + h * 160;
    const float* wi = wj + 64;
    float aj = 0.f, ai = 0.f;
    #pragma unroll 8
    for (int o = 0; o < 64; ++o) { float z = zp[o]; aj += z * wj[o]; ai += z * wi[o]; }
    si[id] = ai; sj[id] = aj;
}

// ---------------------------------------------------------------------------
// Bandwidth-dominant kernel, WMMA edition.
// One block (256 thr = 8 waves) per (b,i) row of e. Each wave computes
// 16-edge x 16-col logit tiles with V_WMMA_F32_16X16X4_F32:
//   cols 0..7  = heads (B fragment = w_e[h][k]),  cols 8..15 = zero pad
//   K = 36: ch 0..31 = e[b,i,j,:] (non-temporal), ch 32 = 1 (A) x s_i[h] (B)
// Post-WMMA per element: + s_j[h][j], leaky-relu, exp, per-head accumulate.
// S[b,h,i] = sum_j exp(lrelu(s_i + s_j + e.w_e))  (shift-invariant softmax)
// ---------------------------------------------------------------------------
__global__ void k_se_rowsum(const float* __restrict__ e, const float* __restrict__ aw,
                            const float* __restrict__ si, const float* __restrict__ sj,
                            float* __restrict__ S)
{
    __shared__ float sj_s[HEADS][NNODE];   // 16 KB
    __shared__ float red[8][32];           // per-wave, per-lane partials

    const int tid  = threadIdx.x;
    const int wave = tid >> 5;
    const int lane = tid & 31;
    const int half = lane >> 4;
    const int lr   = lane & 15;

    const int bi = blockIdx.x;             // b*512 + i
    const int b  = bi >> 9;
    const int i  = bi & 511;

    // stage s_j for all heads of this batch
    for (int k = tid; k < HEADS * NNODE; k += 256) {
        int h = k >> 9, j = k & 511;
        sj_s[h][j] = sj[((b << 3) + h) * NNODE + j];
    }
    __syncthreads();

    // Loop-invariant B fragments: bk[kk] covers K = kk*4 + {2*half, 2*half+1}
    // B[k][col] = col<8 ? (k<32 ? w_e[col][k] : (k==32 ? s_i[col] : 0)) : 0
    v2f bk[9];
    #pragma unroll
    for (int kk = 0; kk < 9; ++kk) {
        float bv[2];
        #pragma unroll
        for (int c = 0; c < 2; ++c) {
            int k = kk * 4 + 2 * half + c;
            float v = 0.f;
            if (lr < 8) {
                if (k < 32)       v = aw[lr * 160 + 128 + k];
                else if (k == 32) v = si[((b << 3) + lr) * NNODE + i];
            }
            bv[c] = v;
        }
        bk[kk].x = bv[0]; bk[kk].y = bv[1];
    }
    // constant A fragment for the augmented channel block (k=32 -> 1, 33..35 -> 0)
    v2f ac; ac.x = (half == 0) ? 1.f : 0.f; ac.y = 0.f;

    float accl = 0.f;                      // this lane: head lr (valid if lr<8)
    for (int jt = wave; jt < 32; jt += 8) {        // 16-edge tiles of j
        const float* ep = e + (((size_t)bi * NNODE) + jt * 16 + lr) * 32 + 2 * half;
        v8f c = {};
        #pragma unroll
        for (int kk = 0; kk < 8; ++kk) {
            v2f a = __builtin_nontemporal_load((const v2f*)(ep + kk * 4));
            c = __builtin_amdgcn_wmma_f32_16x16x4_f32(false, a, false, bk[kk],
                                                      (short)0, c, false, false);
        }
        c = __builtin_amdgcn_wmma_f32_16x16x4_f32(false, ac, false, bk[8],
                                                  (short)0, c, false, false);
        // c[v]: edge = jt*16 + v + 8*half, head = lr (cols 8..15 are pad)
        #pragma unroll
        for (int v = 0; v < 8; ++v) {
            int j = jt * 16 + v + 8 * half;
            float l = c[v] + sj_s[lr & 7][j];
            l = (l >= 0.f) ? l : 0.01f * l;
            accl += (lr < 8) ? __expf(l) : 0.f;
        }
    }

    red[wave][lane] = accl;
    __syncthreads();
    // head h partial lives in lanes {h, h+16} of every wave
    if (tid < HEADS) {
        float t = 0.f;
        #pragma unroll
        for (int w = 0; w < 8; ++w) t += red[w][tid] + red[w][tid + 16];
        S[((b << 3) + tid) * NNODE + i] = t;
    }
}

// total[b,h] = sum_i S[b,h,i]   (grid = 64 blocks, 256 threads)
__global__ void k_total(const float* __restrict__ S, float* __restrict__ tot)
{
    __shared__ float r[256];
    int bh = blockIdx.x, tid = threadIdx.x;
    r[tid] = S[bh * NNODE + tid] + S[bh * NNODE + tid + 256];
    __syncthreads();
    for (int s = 128; s > 0; s >>= 1) {
        if (tid < s) r[tid] += r[tid + s];
        __syncthreads();
    }
    if (tid == 0) tot[bh] = r[0];
}

// merge 'cat': Hn[bn, h*64+o] = mask * sigmoid(z * rowfrac)
__global__ void k_merge_cat(const float* __restrict__ Z, const float* __restrict__ S,
                            const float* __restrict__ tot, const float* __restrict__ mask,
                            float* __restrict__ Hn)
{
    int id = blockIdx.x * 256 + threadIdx.x;
    if (id >= BN * 512) return;
    int c = id & 511, bn = id >> 9;
    int h = c >> 6, b = bn >> 9, n = bn & 511;
    float frac = S[((b << 3) + h) * NNODE + n] / tot[(b << 3) + h];
    Hn[id] = sigf(Z[id] * frac) * mask[bn];
}

// merge 'mean': Hn[bn, o] = mask * sigmoid(mean_h(z * rowfrac))
__global__ void k_merge_mean(const float* __restrict__ Z, const float* __restrict__ S,
                             const float* __restrict__ tot, const float* __restrict__ mask,
                             float* __restrict__ Hn)
{
    int id = blockIdx.x * 256 + threadIdx.x;
    if (id >= BN * HID) return;
    int o = id & 63, bn = id >> 6;
    int b = bn >> 9, n = bn & 511;
    float a = 0.f;
    #pragma unroll
    for (int h = 0; h < HEADS; ++h) {
        float frac = S[((b << 3) + h) * NNODE + n] / tot[(b << 3) + h];
        a += Z[(size_t)bn * 512 + h * 64 + o] * frac;
    }
    a *= 0.125f;
    Hn[id] = sigf(a) * mask[bn];
}

// x = concat(h0, h3) along feature
__global__ void k_cat(const float* __restrict__ h0, const float* __restrict__ h3,
                      float* __restrict__ x)
{
    int id = blockIdx.x * 256 + threadIdx.x;
    if (id >= BN * 128) return;
    int c = id & 127, bn = id >> 7;
    x[id] = (c < 64) ? h0[bn * 64 + c] : h3[bn * 64 + (c - 64)];
}

// out[b,l] = sum_n gate[b,n,l] * val[b,n,l]   (grid = 96 blocks of (b,l))
__global__ void k_out(const float* __restrict__ gate, const float* __restrict__ val,
                      float* __restrict__ out)
{
    __shared__ float r[128];
    int b = blockIdx.x / 12, l = blockIdx.x % 12;
    int tid = threadIdx.x;
    float a = 0.f;
    for (int n = tid; n < NNODE; n += 128) {
        size_t idx = ((size_t)(b * NNODE + n)) * 12 + l;
        a += gate[idx] * val[idx];
    }
    r[tid] = a; __syncthreads();
    for (int s = 64; s > 0; s >>= 1) {
        if (tid < s) r[tid] += r[tid + s];
        __syncthreads();
    }
    if (tid == 0) out[blockIdx.x] = r[0];
}

// ---------------------------------------------------------------------------
extern "C" void kernel_launch(void* const* d_in, const int* in_sizes, int n_in,
                              void* d_out, int out_size, void* d_ws, size_t ws_size,
                              hipStream_t stream)
{
    (void)in_sizes; (void)n_in; (void)out_size; (void)ws_size;

    // setup_inputs() insertion order:
    // 0:g 1:h_in 2:e 3:fc_w0 4:attn_w0 5:fc_w1 6:attn_w1 7:fc_w2 8:attn_w2
    // 9-12:i_ws 13-16:i_bs 17-20:j_ws 21-24:j_bs
    const float* h_in = (const float*)d_in[1];
    const float* e    = (const float*)d_in[2];
    const float* fcw[3]  = {(const float*)d_in[3], (const float*)d_in[5], (const float*)d_in[7]};
    const float* aw[3]   = {(const float*)d_in[4], (const float*)d_in[6], (const float*)d_in[8]};
    const float* iW[4]   = {(const float*)d_in[9],  (const float*)d_in[10], (const float*)d_in[11], (const float*)d_in[12]};
    const float* iB[4]   = {(const float*)d_in[13], (const float*)d_in[14], (const float*)d_in[15], (const float*)d_in[16]};
    const float* jW[4]   = {(const float*)d_in[17], (const float*)d_in[18], (const float*)d_in[19], (const float*)d_in[20]};
    const float* jB[4]   = {(const float*)d_in[21], (const float*)d_in[22], (const float*)d_in[23], (const float*)d_in[24]};
    float* out = (float*)d_out;

    // workspace layout (floats)
    float* ws   = (float*)d_ws;
    float* h0   = ws;                    // 4096*64
    float* mask = h0   + BN * HID;       // 4096
    float* Z    = mask + BN;             // 4096*512
    float* hA   = Z    + BN * 512;       // 4096*512
    float* hB   = hA   + BN * 512;       // 4096*512
    float* si   = hB   + BN * 512;       // 32768
    float* sj   = si   + BATCH*HEADS*NNODE;
    float* Ssum = sj   + BATCH*HEADS*NNODE;
    float* tot  = Ssum + BATCH*HEADS*NNODE;  // 64
    // reuse after the attention layers:
    float* xcat = Z;                     // 4096*128 (Z is free post-layers)
    float* buf0 = Z  + BN * 128;         // 4096*256 (fits inside Z)
    float* buf1 = hA;                    // 4096*256 (hA free after layer 2)
    float* gate = hA + BN * 256;         // 4096*12
    float* valb = gate + BN * 12;        // 4096*12

    k_h0<<<(BN * HID + 255) / 256, 256, 0, stream>>>(h_in, h0, mask);

    for (int t = 0; t < 3; ++t) {
        const float* Xp = (t == 0) ? h0 : (t == 1 ? hB : hA);
        int K = (t == 0) ? 64 : 512;

        // z[bn, h*64+o] : per-head GEMM via f32 WMMA (grid.z = head)
        k_gemm_wmma<<<dim3(BN / 16, 4, HEADS), 32, 0, stream>>>(
            Xp, fcw[t], nullptr, Z, BN, 64, K, 512, 0, (long)64 * K, 64, 0);

        k_sij<<<(BATCH*HEADS*NNODE + 255) / 256, 256, 0, stream>>>(Z, aw[t], si, sj);
        k_se_rowsum<<<BN, 256, 0, stream>>>(e, aw[t], si, sj, Ssum);
        k_total<<<BATCH * HEADS, 256, 0, stream>>>(Ssum, tot);

        if (t < 2) {
            float* Hn = (t == 0) ? hB : hA;
            k_merge_cat<<<(BN * 512 + 255) / 256, 256, 0, stream>>>(Z, Ssum, tot, mask, Hn);
        } else {
            k_merge_mean<<<(BN * HID + 255) / 256, 256, 0, stream>>>(Z, Ssum, tot, mask, hB);
        }
    }

    // readout: gate = sigmoid(MLP([h0, h3])), val = MLP(h3), out = sum_n gate*val
    k_cat<<<(BN * 128 + 255) / 256, 256, 0, stream>>>(h0, hB, xcat);

    k_gemm_wmma<<<dim3(BN/16,  8, 1), 32, 0, stream>>>(xcat, iW[0], iB[0], buf0, BN, 128, 128, 128, 0, 0, 0, 1);
    k_gemm_wmma<<<dim3(BN/16, 16, 1), 32, 0, stream>>>(buf0, iW[1], iB[1], buf1, BN, 256, 128, 256, 0, 0, 0, 1);
    k_gemm_wmma<<<dim3(BN/16,  8, 1), 32, 0, stream>>>(buf1, iW[2], iB[2], buf0, BN, 128, 256, 128, 0, 0, 0, 1);
    k_gemm_wmma<<<dim3(BN/16,  1, 1), 32, 0, stream>>>(buf0, iW[3], iB[3], gate, BN,  12, 128,  12, 0, 0, 0, 2);

    k_gemm_wmma<<<dim3(BN/16,  8, 1), 32, 0, stream>>>(hB,   jW[0], jB[0], buf0, BN, 128,  64, 128, 0, 0, 0, 1);
    k_gemm_wmma<<<dim3(BN/16, 16, 1), 32, 0, stream>>>(buf0, jW[1], jB[1], buf1, BN, 256, 128, 256, 0, 0, 0, 1);
    k_gemm_wmma<<<dim3(BN/16,  8, 1), 32, 0, stream>>>(buf1, jW[2], jB[2], buf0, BN, 128, 256, 128, 0, 0, 0, 1);
    k_gemm_wmma<<<dim3(BN/16,  1, 1), 32, 0, stream>>>(buf0, jW[3], jB[3], valb, BN,  12, 128,  12, 0, 0, 0, 0);

    k_out<<<BATCH * 12, 128, 0, stream>>>(gate, valb, out);
}